// TFNConv_13958643712280
// MI455X (gfx1250) — compile-verified
//
#include <hip/hip_runtime.h>

// ---------------------------------------------------------------------------
// TFN conv layer for MI455X (gfx1250, wave32).
//  - Dense channel mixes on V_WMMA_F32_16X16X4_F32, one 16-node tile per wave
//    (no grid-stride loop: avoids LICM hoisting LDS weight loads -> spills).
//  - Weights staged in LDS in pair-packed layout {W[k][n],W[k+1][n]} so each
//    WMMA B-operand is a single aligned ds_load_b64.
//  - Vector GEMMs process all 3 Cartesian components per k-step (shared B).
//  - Edge scatter: relaxed agent-scope f32 atomics -> global_atomic_add_f32
//    resolving in the 192MB L2 (agg buffers are 70MB -> L2-resident).
// ---------------------------------------------------------------------------

typedef float v2f __attribute__((ext_vector_type(2)));
typedef float v8f __attribute__((ext_vector_type(8)));

#define INV_SQRT32 0.17677669529663689f
#define INV_SQRT8  0.35355339059327379f
#define CG_INV3    0.57735026918962584f   // 1/sqrt(3)
#define CG_INV2    0.70710678118654752f   // 1/sqrt(2)

__device__ __forceinline__ v8f wmma4(v2f a, v2f b, v8f c) {
  // D = A(16x4) * B(4x16) + C(16x16), all f32.
  return __builtin_amdgcn_wmma_f32_16x16x4_f32(false, a, false, b, (short)0, c,
                                               false, false);
}

__device__ __forceinline__ void atomic_add_f32(float* p, float v) {
  // no-return relaxed agent-scope -> hardware global_atomic_add_f32 (L2 RMW)
  __hip_atomic_fetch_add(p, v, __ATOMIC_RELAXED, __HIP_MEMORY_SCOPE_AGENT);
}

// Stage a [rows,32] row-major weight panel into LDS, pair-packed:
//   s[((k>>1)*32 + n)*2 + (k&1)] = W[k][n]
// so lane reads {W[k][n], W[k+1][n]} (k even) as one b64 at pair index k>>1.
__device__ __forceinline__ void stage_pairs(const float* __restrict__ g,
                                            float* __restrict__ s,
                                            int rows, int tid, int nthreads) {
  for (int i = tid; i < rows * 32; i += nthreads) {
    const int row = i >> 5, col = i & 31;
    s[(((row >> 1) * 32 + col) << 1) | (row & 1)] = g[i];
  }
}

__device__ __forceinline__ v2f lds_pair(const float* __restrict__ s,
                                        int pair, int col) {
  return *(const v2f*)&s[(pair * 32 + col) * 2];
}

// ---------------------------------------------------------------------------
__global__ __launch_bounds__(256) void zero_kernel(float* __restrict__ p, size_t n) {
  size_t i = (size_t)blockIdx.x * blockDim.x + threadIdx.x;
  size_t stride = (size_t)gridDim.x * blockDim.x;
  for (; i < n; i += stride) p[i] = 0.0f;
}

// ---------------------------------------------------------------------------
// linear_1: x_s = node_s @ W1_s / sqrt(32);  x_v[:,:,c] = node_v[:,:,c] @ W1_v / sqrt(32)
// One wave per 16-node tile; tiles*16 == N (N % 16 == 0) -> full EXEC for WMMA.
__global__ __launch_bounds__(128) void linear1_kernel(
    const float* __restrict__ node_s, const float* __restrict__ node_v,
    const float* __restrict__ W1_s, const float* __restrict__ W1_v,
    float* __restrict__ x_s, float* __restrict__ x_v, int tiles) {
  __shared__ float sWs[1024];   // [16 pairs][32 cols][2]
  __shared__ float sWv[1024];
  stage_pairs(W1_s, sWs, 32, threadIdx.x, blockDim.x);
  stage_pairs(W1_v, sWv, 32, threadIdx.x, blockDim.x);
  __syncthreads();

  const int lane = threadIdx.x & 31;
  const int tile = blockIdx.x * (blockDim.x >> 5) + (threadIdx.x >> 5);
  if (tile >= tiles) return;             // wave-uniform exit
  const int m    = lane & 15;            // A row / C column within tile
  const int hi   = lane >> 4;            // half-wave: selects K pair
  const int row0 = tile * 16;
  const int r    = row0 + m;

  // ---- scalars ----
  for (int n0 = 0; n0 < 32; n0 += 16) {
    v8f acc = {};
#pragma unroll 4
    for (int k0 = 0; k0 < 32; k0 += 4) {
      const int k = k0 + hi * 2;
      const v2f a = *(const v2f*)&node_s[r * 32 + k];
      const v2f b = lds_pair(sWs, (k0 >> 1) + hi, n0 + m);
      acc = wmma4(a, b, acc);
    }
#pragma unroll
    for (int rr = 0; rr < 8; ++rr)
      x_s[(row0 + rr + 8 * hi) * 32 + n0 + m] = acc[rr] * INV_SQRT32;
  }

  // ---- vectors: 3 components per k-step (shared B, contiguous A) ----
  for (int n0 = 0; n0 < 32; n0 += 16) {
    v8f a0 = {}, a1 = {}, a2 = {};
#pragma unroll 4
    for (int k0 = 0; k0 < 32; k0 += 4) {
      const int k = k0 + hi * 2;
      const v2f b = lds_pair(sWv, (k0 >> 1) + hi, n0 + m);
      const float* ap = &node_v[(r * 32 + k) * 3];   // 6 contiguous floats
      v2f ax, ay, az;
      ax.x = ap[0]; ay.x = ap[1]; az.x = ap[2];
      ax.y = ap[3]; ay.y = ap[4]; az.y = ap[5];
      a0 = wmma4(ax, b, a0);
      a1 = wmma4(ay, b, a1);
      a2 = wmma4(az, b, a2);
    }
#pragma unroll
    for (int rr = 0; rr < 8; ++rr) {
      float* op = &x_v[(row0 + rr + 8 * hi) * 96 + (n0 + m) * 3];  // b96 store
      op[0] = a0[rr] * INV_SQRT32;
      op[1] = a1[rr] * INV_SQRT32;
      op[2] = a2[rr] * INV_SQRT32;
    }
  }
}

// ---------------------------------------------------------------------------
// Edge kernel: one wave per edge, lane == channel u (MUL=32 == wave32).
// Gather -> radial MLP -> 5 CG paths -> 11 f32 atomics per lane into L2.
__global__ __launch_bounds__(256) void edge_kernel(
    const float* __restrict__ x_s, const float* __restrict__ x_v,
    const float* __restrict__ edge_emb, const float* __restrict__ edge_sh0,
    const float* __restrict__ edge_sh1, const float* __restrict__ Wm1,
    const float* __restrict__ Wm2, const int* __restrict__ edge_src,
    const int* __restrict__ edge_dst, float* __restrict__ agg_s,
    float* __restrict__ agg_v, int E) {
  __shared__ float sWm1[64];     // [8,8]
  __shared__ float sWm2[1280];   // [8,160]
  for (int i = threadIdx.x; i < 64; i += blockDim.x) sWm1[i] = Wm1[i];
  for (int i = threadIdx.x; i < 1280; i += blockDim.x) sWm2[i] = Wm2[i];
  __syncthreads();

  const int lane = threadIdx.x & 31;
  const int e = blockIdx.x * (blockDim.x >> 5) + (threadIdx.x >> 5);
  if (e >= E) return;

  const int src = edge_src[e];
  const int dst = edge_dst[e];
  const float s0  = edge_sh0[e];
  const float shx = edge_sh1[e * 3 + 0];
  const float shy = edge_sh1[e * 3 + 1];
  const float shz = edge_sh1[e * 3 + 2];

  // radial MLP: h = silu(emb @ Wm1 / sqrt(8)); w = h @ Wm2 / sqrt(8)
  float emb[8];
#pragma unroll
  for (int i = 0; i < 8; ++i) emb[i] = edge_emb[e * 8 + i];
  float h[8];
#pragma unroll
  for (int j = 0; j < 8; ++j) {
    float acc = 0.0f;
#pragma unroll
    for (int i = 0; i < 8; ++i) acc += emb[i] * sWm1[i * 8 + j];
    acc *= INV_SQRT8;
    h[j] = acc / (1.0f + __expf(-acc));   // silu
  }
  float w[5];
#pragma unroll
  for (int p = 0; p < 5; ++p) {
    float acc = 0.0f;
#pragma unroll
    for (int j = 0; j < 8; ++j) acc += h[j] * sWm2[j * 160 + p * 32 + lane];
    w[p] = acc * INV_SQRT8;
  }

  // gather source features (x_s + x_v = 26MB -> L2-resident)
  const float su = x_s[src * 32 + lane];
  const float* vp = &x_v[src * 96 + lane * 3];   // b96 gather
  const float vx = vp[0], vy = vp[1], vz = vp[2];

  // Clebsch-Gordan paths (lmax = 1)
  const float p1  = w[0] * su * s0;                               // 0e x 0e -> 0e
  const float dot = vx * shx + vy * shy + vz * shz;
  const float p2  = w[1] * dot * CG_INV3;                         // 1o x 1o -> 0e
  const float p3x = w[2] * su * shx, p3y = w[2] * su * shy, p3z = w[2] * su * shz;
  const float p4x = w[3] * vx * s0,  p4y = w[3] * vy * s0,  p4z = w[3] * vz * s0;
  const float cxp = vy * shz - vz * shy;                          // v x sh1
  const float cyp = vz * shx - vx * shz;
  const float czp = vx * shy - vy * shx;
  const float p5x = w[4] * cxp * CG_INV2, p5y = w[4] * cyp * CG_INV2,
              p5z = w[4] * czp * CG_INV2;

  // scatter-add (agg buffers fit in L2; relaxed agent-scope f32 atomics)
  atomic_add_f32(&agg_s[(size_t)dst * 64 + lane], p1);
  atomic_add_f32(&agg_s[(size_t)dst * 64 + 32 + lane], p2);
  float* av = &agg_v[(size_t)dst * 288];
  atomic_add_f32(&av[lane * 3 + 0], p3x);
  atomic_add_f32(&av[lane * 3 + 1], p3y);
  atomic_add_f32(&av[lane * 3 + 2], p3z);
  atomic_add_f32(&av[(32 + lane) * 3 + 0], p4x);
  atomic_add_f32(&av[(32 + lane) * 3 + 1], p4y);
  atomic_add_f32(&av[(32 + lane) * 3 + 2], p4z);
  atomic_add_f32(&av[(64 + lane) * 3 + 0], p5x);
  atomic_add_f32(&av[(64 + lane) * 3 + 1], p5y);
  atomic_add_f32(&av[(64 + lane) * 3 + 2], p5z);
}

// ---------------------------------------------------------------------------
// finalize: linear_2 + self-connection fused, one 16-node tile per wave.
// Weight panels pair-packed in 84KB dynamic LDS (WGP has 320KB).
// Self-connection A = outer(node_feat, attrs): k = u*8 + a, built on the fly.
__global__ __launch_bounds__(128) void finalize_kernel(
    const float* __restrict__ node_s, const float* __restrict__ node_v,
    const float* __restrict__ attrs, const float* __restrict__ agg_s,
    const float* __restrict__ agg_v, const float* __restrict__ W2_s,
    const float* __restrict__ W2_v, const float* __restrict__ Wsc_s,
    const float* __restrict__ Wsc_v, float* __restrict__ out, int tiles) {
  extern __shared__ float smem[];
  float* sW2s = smem;            // [64,32]  pair-packed = 2048 floats
  float* sW2v = smem + 2048;     // [96,32]  = 3072
  float* sWss = smem + 5120;     // [256,32] = 8192
  float* sWsv = smem + 13312;    // [256,32] = 8192   (total 21504 floats = 84KB)
  stage_pairs(W2_s, sW2s, 64, threadIdx.x, blockDim.x);
  stage_pairs(W2_v, sW2v, 96, threadIdx.x, blockDim.x);
  stage_pairs(Wsc_s, sWss, 256, threadIdx.x, blockDim.x);
  stage_pairs(Wsc_v, sWsv, 256, threadIdx.x, blockDim.x);
  __syncthreads();

  const int lane = threadIdx.x & 31;
  const int tile = blockIdx.x * (blockDim.x >> 5) + (threadIdx.x >> 5);
  if (tile >= tiles) return;
  const int m    = lane & 15;
  const int hi   = lane >> 4;
  const int row0 = tile * 16;
  const int r    = row0 + m;

  const float SC_MS = 0.03125f;          // (1/sqrt(16)) * (1/sqrt(64))
  const float SC_MV = 0.025515518f;      // (1/sqrt(16)) * (1/sqrt(96))
  const float SC_SC = 0.0625f;           // 1/sqrt(32*8)

  // ---- scalar output: out[:, 0:32] ----
  for (int n0 = 0; n0 < 32; n0 += 16) {
    v8f accL = {};                                  // agg_s @ W2_s   (K=64)
#pragma unroll 4
    for (int k0 = 0; k0 < 64; k0 += 4) {
      const int k = k0 + hi * 2;
      const v2f a = *(const v2f*)&agg_s[(size_t)r * 64 + k];
      const v2f b = lds_pair(sW2s, (k0 >> 1) + hi, n0 + m);
      accL = wmma4(a, b, accL);
    }
    v8f accS = {};                                  // outer(s,attr) @ Wsc_s (K=256)
#pragma unroll 4
    for (int k0 = 0; k0 < 256; k0 += 4) {
      const int k = k0 + hi * 2;
      v2f a;
      a.x = node_s[r * 32 + (k >> 3)]       * attrs[r * 8 + (k & 7)];
      a.y = node_s[r * 32 + ((k + 1) >> 3)] * attrs[r * 8 + ((k + 1) & 7)];
      const v2f b = lds_pair(sWss, (k0 >> 1) + hi, n0 + m);
      accS = wmma4(a, b, accS);
    }
#pragma unroll
    for (int rr = 0; rr < 8; ++rr)
      out[(row0 + rr + 8 * hi) * 128 + n0 + m] = accL[rr] * SC_MS + accS[rr] * SC_SC;
  }

  // ---- vector outputs: out[:, 32 + u*3 + c], 3 components per k-step ----
  for (int n0 = 0; n0 < 32; n0 += 16) {
    v8f l0 = {}, l1 = {}, l2 = {};                  // agg_v @ W2_v (K=96)
#pragma unroll 4
    for (int k0 = 0; k0 < 96; k0 += 4) {
      const int k = k0 + hi * 2;
      const v2f b = lds_pair(sW2v, (k0 >> 1) + hi, n0 + m);
      const float* ap = &agg_v[(size_t)r * 288 + k * 3];  // 6 contiguous floats
      v2f ax, ay, az;
      ax.x = ap[0]; ay.x = ap[1]; az.x = ap[2];
      ax.y = ap[3]; ay.y = ap[4]; az.y = ap[5];
      l0 = wmma4(ax, b, l0);
      l1 = wmma4(ay, b, l1);
      l2 = wmma4(az, b, l2);
    }
    v8f s0a = {}, s1a = {}, s2a = {};               // outer(v_c,attr) @ Wsc_v (K=256)
#pragma unroll 4
    for (int k0 = 0; k0 < 256; k0 += 4) {
      const int k = k0 + hi * 2;
      const v2f b = lds_pair(sWsv, (k0 >> 1) + hi, n0 + m);
      const float at0 = attrs[r * 8 + (k & 7)];
      const float at1 = attrs[r * 8 + ((k + 1) & 7)];
      const float* v0 = &node_v[(r * 32 + (k >> 3)) * 3];
      const float* v1 = &node_v[(r * 32 + ((k + 1) >> 3)) * 3];
      v2f ax, ay, az;
      ax.x = v0[0] * at0; ay.x = v0[1] * at0; az.x = v0[2] * at0;
      ax.y = v1[0] * at1; ay.y = v1[1] * at1; az.y = v1[2] * at1;
      s0a = wmma4(ax, b, s0a);
      s1a = wmma4(ay, b, s1a);
      s2a = wmma4(az, b, s2a);
    }
#pragma unroll
    for (int rr = 0; rr < 8; ++rr) {
      float* op = &out[(row0 + rr + 8 * hi) * 128 + 32 + (n0 + m) * 3];  // b96
      op[0] = l0[rr] * SC_MV + s0a[rr] * SC_SC;
      op[1] = l1[rr] * SC_MV + s1a[rr] * SC_SC;
      op[2] = l2[rr] * SC_MV + s2a[rr] * SC_SC;
    }
  }
}

// ---------------------------------------------------------------------------
extern "C" void kernel_launch(void* const* d_in, const int* in_sizes, int n_in,
                              void* d_out, int out_size, void* d_ws, size_t ws_size,
                              hipStream_t stream) {
  const float* node_s   = (const float*)d_in[0];
  const float* node_v   = (const float*)d_in[1];
  const float* attrs    = (const float*)d_in[2];
  const float* edge_emb = (const float*)d_in[3];
  const float* edge_sh0 = (const float*)d_in[4];
  const float* edge_sh1 = (const float*)d_in[5];
  const float* W1_s     = (const float*)d_in[6];
  const float* W1_v     = (const float*)d_in[7];
  const float* Wm1      = (const float*)d_in[8];
  const float* Wm2      = (const float*)d_in[9];
  const float* W2_s     = (const float*)d_in[10];
  const float* W2_v     = (const float*)d_in[11];
  const float* Wsc_s    = (const float*)d_in[12];
  const float* Wsc_v    = (const float*)d_in[13];
  const int*   edge_src = (const int*)d_in[14];
  const int*   edge_dst = (const int*)d_in[15];
  float* out = (float*)d_out;

  const int N = in_sizes[0] / 32;   // node_s is [N, 32]
  const int E = in_sizes[3] / 8;    // edge_emb is [E, 8]
  const int tiles = (N + 15) / 16;  // N % 16 == 0 for this problem

  // workspace layout (floats): x_s[N*32] | x_v[N*96] | agg_s[N*64] | agg_v[N*288]
  float* ws    = (float*)d_ws;
  float* x_s   = ws;
  float* x_v   = ws + (size_t)N * 32;
  float* agg_s = ws + (size_t)N * 128;
  float* agg_v = ws + (size_t)N * 192;

  // 1) zero aggregation buffers (every call: harness poisons d_ws)
  zero_kernel<<<2048, 256, 0, stream>>>(agg_s, (size_t)N * 352);

  // 2) linear_1 via f32 WMMA, pair-packed W1 in LDS, one tile per wave
  linear1_kernel<<<(tiles + 3) / 4, 128, 0, stream>>>(node_s, node_v, W1_s, W1_v,
                                                      x_s, x_v, tiles);

  // 3) edge messages + scatter (L2-resident atomics); one wave per edge
  edge_kernel<<<(E + 7) / 8, 256, 0, stream>>>(x_s, x_v, edge_emb, edge_sh0,
                                               edge_sh1, Wm1, Wm2, edge_src,
                                               edge_dst, agg_s, agg_v, E);

  // 4) linear_2 + self-connection via f32 WMMA, weights in 84KB dynamic LDS
  finalize_kernel<<<(tiles + 3) / 4, 128, 21504 * sizeof(float), stream>>>(
      node_s, node_v, attrs, agg_s, agg_v, W2_s, W2_v, Wsc_s, Wsc_v, out, tiles);
}